// GraphSoftmax_48902497632441
// MI455X (gfx1250) — compile-verified
//
#include <hip/hip_runtime.h>
#include <hip/hip_bf16.h>
#include <stdint.h>

// Segment-wise softmax over sorted batch ids.
//   d_in[0]: batch  (int64, N)       -- sorted, values in [0, 8192)
//   d_in[1]: input  (float32, N x 32)
//   d_out  : output (float32, N x 32)
//
// One workgroup per segment. Segments are contiguous (batch sorted), so the
// Tensor Data Mover DMAs the whole segment slab into LDS once; all three
// softmax passes then run out of LDS. HBM traffic = 1 read + 1 write (the
// bandwidth floor: 512 MB -> ~22us at 23.3 TB/s).

#define NSEG     8192
#define TPB      256      // 8 wave32s
#define CAP_ROWS 480      // 480*128B = 60KB LDS tile (mean seg = 244 rows)

typedef __attribute__((ext_vector_type(4))) unsigned int v4u;
typedef __attribute__((ext_vector_type(8))) int          v8i;
typedef __attribute__((ext_vector_type(4))) int          v4i;

__device__ __forceinline__ unsigned rfl(unsigned x) {
  return (unsigned)__builtin_amdgcn_readfirstlane((int)x);
}

__global__ __launch_bounds__(TPB) void GraphSoftmax_48902497632441_kernel(
    const long long* __restrict__ batch, const float* __restrict__ x,
    float* __restrict__ out, int N)
{
  __shared__ int   sbound[2];
  __shared__ float smax[32];
  __shared__ float sinv[32];
  __shared__ float red[8][32];
  __shared__ __align__(16) float sx[CAP_ROWS * 32];

  const int tid = threadIdx.x;
  const int seg = blockIdx.x;

  // Segment bounds via binary search on sorted batch (tid 0: lower_bound(seg),
  // tid 1: lower_bound(seg+1)).
  if (tid < 2) {
    long long key = (long long)seg + (long long)tid;
    int lo = 0, hi = N;
    while (lo < hi) {
      int mid = (lo + hi) >> 1;
      if (batch[mid] < key) lo = mid + 1; else hi = mid;
    }
    sbound[tid] = lo;
  }
  __syncthreads();
  const int start = sbound[0];
  const int rows  = sbound[1] - start;
  if (rows <= 0) return;   // uniform across block

  const int f  = tid & 31;   // feature owned by this lane
  const int rg = tid >> 5;   // row-group (wave id)

  if (rows <= CAP_ROWS) {
    // ---- TDM path: DMA the contiguous [rows x 32] f32 slab into LDS ----
    if (tid < 32) {  // wave 0 issues one TDM op; EXEC is ignored by TDM
      const unsigned elems = (unsigned)rows * 32u;           // <= 15360
      const unsigned long long ga =
          (unsigned long long)(uintptr_t)(x + (size_t)start * 32);
      const unsigned lds = (unsigned)(uintptr_t)(&sx[0]);    // LDS byte offset

      // D# group 0: count=1 (valid, load, user), lds_addr, global_addr, type=2
      v4u g0;
      g0[0] = rfl(1u);
      g0[1] = rfl(lds);
      g0[2] = rfl((unsigned)(ga & 0xFFFFFFFFull));
      g0[3] = rfl(((unsigned)(ga >> 32) & 0x01FFFFFFu) | (2u << 30));
      // D# group 1: data_size=4B; 1-D tile: tile_dim0 = tensor_dim0 = elems
      v8i g1;
      g1[0] = (int)rfl(2u << 16);                    // data_size = 2 (4 bytes)
      g1[1] = (int)rfl((elems & 0xFFFFu) << 16);     // tensor_dim0[15:0] @ [63:48]
      g1[2] = (int)rfl((elems >> 16) | (1u << 16));  // tensor_dim0[31:16] | tensor_dim1=1
      g1[3] = (int)rfl(elems << 16);                 // tile_dim0 @ [127:112]
      g1[4] = (int)rfl(0u);                          // tile_dim1=0, tile_dim2=0
      g1[5] = (int)rfl(elems);                       // tensor_dim0_stride
      g1[6] = (int)rfl(0u);
      g1[7] = (int)rfl(0u);
      v4i g2, g3;                                    // unused dims: zero (2-D form)
      g2[0] = g2[1] = g2[2] = g2[3] = 0;
      g3[0] = g3[1] = g3[2] = g3[3] = 0;
      v8i g4;                                        // extra group (clang-23 form): zero
      g4[0] = g4[1] = g4[2] = g4[3] = 0;
      g4[4] = g4[5] = g4[6] = g4[7] = 0;

      __builtin_amdgcn_tensor_load_to_lds(g0, g1, g2, g3, g4, 0);
      __builtin_amdgcn_s_wait_tensorcnt(0);
    }
    __syncthreads();  // LDS tile visible to all 8 waves

    // ---- pass 1: per-feature max ----
    float m = -3.402823466e38f;
    for (int r = rg; r < rows; r += 8) m = fmaxf(m, sx[r * 32 + f]);
    red[rg][f] = m;
    __syncthreads();
    if (rg == 0) {
      float mm = red[0][f];
#pragma unroll
      for (int i = 1; i < 8; ++i) mm = fmaxf(mm, red[i][f]);
      smax[f] = mm;
    }
    __syncthreads();

    // ---- pass 2: exp in place + per-feature sum ----
    const float mf = smax[f];
    float ssum = 0.f;
    for (int r = rg; r < rows; r += 8) {
      float v = __expf(sx[r * 32 + f] - mf);
      sx[r * 32 + f] = v;
      ssum += v;
    }
    red[rg][f] = ssum;
    __syncthreads();
    if (rg == 0) {
      float s = 0.f;
#pragma unroll
      for (int i = 0; i < 8; ++i) s += red[i][f];
      sinv[f] = 1.0f / (s + 1e-10f);
    }
    __syncthreads();

    // ---- pass 3: scale + coalesced b128 store ----
    float* gy = out + (size_t)start * 32;
    const int total = rows * 32;
    for (int i = tid * 4; i < total; i += TPB * 4) {
      float4 v = *(const float4*)(&sx[i]);
      const int fb = i & 31;           // i is 4-aligned -> fb in {0,4,...,28}
      v.x *= sinv[fb + 0];
      v.y *= sinv[fb + 1];
      v.z *= sinv[fb + 2];
      v.w *= sinv[fb + 3];
      *(float4*)(gy + i) = v;
    }
  } else {
    // ---- streaming fallback (statistically never taken): 3 global passes ----
    const float* gx = x + (size_t)start * 32;
    float*       gy = out + (size_t)start * 32;
    float m = -3.402823466e38f;
    for (int r = rg; r < rows; r += 8) m = fmaxf(m, gx[(size_t)r * 32 + f]);
    red[rg][f] = m;
    __syncthreads();
    if (rg == 0) {
      float mm = red[0][f];
#pragma unroll
      for (int i = 1; i < 8; ++i) mm = fmaxf(mm, red[i][f]);
      smax[f] = mm;
    }
    __syncthreads();
    const float mf = smax[f];
    float ssum = 0.f;
    for (int r = rg; r < rows; r += 8) ssum += __expf(gx[(size_t)r * 32 + f] - mf);
    red[rg][f] = ssum;
    __syncthreads();
    if (rg == 0) {
      float s = 0.f;
#pragma unroll
      for (int i = 0; i < 8; ++i) s += red[i][f];
      sinv[f] = 1.0f / (s + 1e-10f);
    }
    __syncthreads();
    const float inv = sinv[f];
    for (int r = rg; r < rows; r += 8)
      gy[(size_t)r * 32 + f] = __expf(gx[(size_t)r * 32 + f] - mf) * inv;
  }
}

extern "C" void kernel_launch(void* const* d_in, const int* in_sizes, int n_in,
                              void* d_out, int out_size, void* d_ws, size_t ws_size,
                              hipStream_t stream) {
  const long long* batch = (const long long*)d_in[0];  // int64 batch ids
  const float*     xin   = (const float*)d_in[1];      // [N, 32] f32
  float*           yout  = (float*)d_out;              // [N, 32] f32
  const int N = in_sizes[0];
  GraphSoftmax_48902497632441_kernel<<<NSEG, TPB, 0, stream>>>(batch, xin, yout, N);
}